// QuantizationLayer_48112223649804
// MI455X (gfx1250) — compile-verified
//
#include <hip/hip_runtime.h>
#include <cstdint>

// ---------------------------------------------------------------------------
// Successive-approximation quantizer (8-bit), LENGTH*PACKET = 8.39M elements.
// Bandwidth-bound: 4B in + 36B out per element (~336 MB -> ~14.4us @ 23.3TB/s).
// No matmul -> WMMA inapplicable. CDNA5 path used: TDM async tensor load to
// LDS (tensor_load_to_lds) double-buffered with s_wait_tensorcnt.
// ---------------------------------------------------------------------------

#define NBITS     8
#define TILE      4096          // elements per TDM tile = 16 KB of f32
#define TPB_TILES 2             // tiles per block, double buffered
#define BLOCK     128           // 4 wave32s
#define GROUPS    (TILE / (BLOCK * 4))   // 8 float4-groups per thread per tile

typedef unsigned int u32x4 __attribute__((ext_vector_type(4)));
typedef int          i32x4 __attribute__((ext_vector_type(4)));
typedef int          i32x8 __attribute__((ext_vector_type(8)));

#if defined(__gfx1250__) && __has_builtin(__builtin_amdgcn_tensor_load_to_lds) && \
    __has_builtin(__builtin_amdgcn_s_wait_tensorcnt)
#define HAVE_TDM 1
#else
#define HAVE_TDM 0
#endif

#if HAVE_TDM
// Build a 1D Tensor-DMA descriptor (D#) per CDNA5 ISA ch.8 and issue the load.
// Group0: [1:0]=count=1, [63:32]=lds_addr, [120:64]=global_addr, [127:126]=type=2
// Group1: [17:16]=data_size(2 => 4B), tensor_dim0=[79:48], tile_dim0=[127:112],
//         tile_dim1=0 -> 1D tile; strides/mask/pad all zero.
__device__ __forceinline__ void tdm_load_1d(unsigned lds_addr, const void* gptr,
                                            unsigned nelem) {
  unsigned long long ga = (unsigned long long)(uintptr_t)gptr;
  u32x4 g0;
  g0[0] = 1u;                                   // count=1, user descriptor
  g0[1] = lds_addr;                             // LDS byte offset
  g0[2] = (unsigned)(ga & 0xFFFFFFFFu);         // global_addr[31:0]
  g0[3] = (unsigned)(ga >> 32) | (2u << 30);    // global_addr[56:32] | type=2

  i32x8 g1;
  g1[0] = (int)(2u << 16);                      // data_size = 4 bytes
  g1[1] = (int)((nelem & 0xFFFFu) << 16);       // tensor_dim0[15:0]
  g1[2] = (int)((nelem >> 16) & 0xFFFFu);       // tensor_dim0[31:16], tensor_dim1=0
  g1[3] = (int)((nelem & 0xFFFFu) << 16);       // tile_dim0 (16-bit)
  g1[4] = 0;                                    // tile_dim1=0 (1D), tile_dim2=0
  g1[5] = 0;                                    // tensor_dim0_stride (unused, 1D)
  g1[6] = 0;
  g1[7] = 0;

  i32x4 z4 = {0, 0, 0, 0};
#if __clang_major__ >= 23
  i32x8 z8 = {0, 0, 0, 0, 0, 0, 0, 0};
  __builtin_amdgcn_tensor_load_to_lds(g0, g1, z4, z4, z8, 0);
#else
  __builtin_amdgcn_tensor_load_to_lds(g0, g1, z4, z4, 0);
#endif
}
#endif

// Per-element SAR quantizer; replicates the reference's exact FP evaluation
// order (all scale factors are exact powers of two in fp32).
__device__ __forceinline__ float quant_elem(float x, float Qc[NBITS]) {
  const float VR = 0.00390625f;   // 1/256
  const float A  = 10000.0f;
  const float D  = 1e-30f;
#pragma unroll
  for (int j = NBITS - 1; j >= 0; --j) {
    float bs = (float)(1 << j) * VR;
#pragma unroll
    for (int k = j + 1; k < NBITS; ++k)
      bs = bs + ((Qc[k] + 1.0f) * 0.5f) * ((float)(1 << k) * VR);
    Qc[j] = tanhf(A * (x - bs + D));
  }
  float q = 0.0f;
#pragma unroll
  for (int b = 0; b < NBITS; ++b)
    q = q + ((Qc[b] + 1.0f) * 0.5f) * (VR * (float)(1 << b));
  return q;
}

__device__ __forceinline__ void process4(float4 xv, size_t e,
                                         float* __restrict__ q_out,
                                         float* __restrict__ Q_out) {
  float xs[4] = {xv.x, xv.y, xv.z, xv.w};
  float qv[4];
#pragma unroll
  for (int s = 0; s < 4; ++s) {
    float Qc[NBITS];
    qv[s] = quant_elem(xs[s], Qc);
    float4 lo = make_float4(Qc[0], Qc[1], Qc[2], Qc[3]);
    float4 hi = make_float4(Qc[4], Qc[5], Qc[6], Qc[7]);
    float* dst = Q_out + (e + (size_t)s) * NBITS;   // 32B-aligned
    *(float4*)(dst + 0) = lo;
    *(float4*)(dst + 4) = hi;
  }
  *(float4*)(q_out + e) = make_float4(qv[0], qv[1], qv[2], qv[3]);
}

__global__ __launch_bounds__(BLOCK) void quant_main(
    const float* __restrict__ x, float* __restrict__ q_out,
    float* __restrict__ Q_out) {
  const int tid = threadIdx.x;
  const size_t base = (size_t)blockIdx.x * (TILE * TPB_TILES);

#if HAVE_TDM
  __shared__ __align__(16) float buf[2][TILE];
  const unsigned lds0 = (unsigned)(uintptr_t)&buf[0][0];  // LDS byte offset

  // Wave 0 issues both TDM loads (branch skips other waves; TDM ignores EXEC).
  if (tid < 32) {
    tdm_load_1d(lds0, x + base, TILE);
    tdm_load_1d(lds0 + TILE * (unsigned)sizeof(float), x + base + TILE, TILE);
  }
#pragma unroll
  for (int it = 0; it < TPB_TILES; ++it) {
    if (tid < 32) {
      // TDM ops complete in order per wave: <=1 guarantees tile0 is resident.
      if (it + 1 < TPB_TILES) __builtin_amdgcn_s_wait_tensorcnt(1);
      else                    __builtin_amdgcn_s_wait_tensorcnt(0);
    }
    __syncthreads();
    const float* src = buf[it & 1];
    const size_t tbase = base + (size_t)it * TILE;
#pragma unroll 1
    for (int g = 0; g < GROUPS; ++g) {
      const int e4 = (g * BLOCK + tid) * 4;
      float4 xv = *(const float4*)(src + e4);     // ds_load_b128
      process4(xv, tbase + (size_t)e4, q_out, Q_out);
    }
  }
#else
#pragma unroll 1
  for (int it = 0; it < TPB_TILES; ++it) {
    const size_t tbase = base + (size_t)it * TILE;
#pragma unroll 1
    for (int g = 0; g < GROUPS; ++g) {
      const int e4 = (g * BLOCK + tid) * 4;
      float4 xv = *(const float4*)(x + tbase + (size_t)e4);
      process4(xv, tbase + (size_t)e4, q_out, Q_out);
    }
  }
#endif
}

__global__ void quant_tail(const float* __restrict__ x, float* __restrict__ q_out,
                           float* __restrict__ Q_out, unsigned long long start,
                           unsigned long long n) {
  unsigned long long i =
      start + (unsigned long long)blockIdx.x * blockDim.x + threadIdx.x;
  if (i >= n) return;
  float Qc[NBITS];
  float q = quant_elem(x[i], Qc);
  q_out[i] = q;
#pragma unroll
  for (int b = 0; b < NBITS; ++b) Q_out[i * NBITS + b] = Qc[b];
}

__global__ void copy_w(const float* __restrict__ W, float* __restrict__ W_out,
                       int n) {
  int i = threadIdx.x;
  if (i < n) W_out[i] = W[i];
}

extern "C" void kernel_launch(void* const* d_in, const int* in_sizes, int n_in,
                              void* d_out, int out_size, void* d_ws,
                              size_t ws_size, hipStream_t stream) {
  (void)d_ws; (void)ws_size; (void)out_size; (void)n_in;
  const float* x = (const float*)d_in[0];
  const float* W = (const float*)d_in[1];
  const size_t nElem = (size_t)in_sizes[0];        // LENGTH*PACKET = 8388608
  const int    nW    = in_sizes[1];                // 36

  float* q_out = (float*)d_out;                    // [nElem]
  float* Q_out = q_out + nElem;                    // [nElem*8]
  float* W_out = Q_out + nElem * NBITS;            // [36]

  const size_t perBlock   = (size_t)TILE * TPB_TILES;   // 8192 elements
  const size_t fullBlocks = nElem / perBlock;           // 1024 for ref sizes
  if (fullBlocks)
    quant_main<<<(unsigned)fullBlocks, BLOCK, 0, stream>>>(x, q_out, Q_out);

  const size_t done = fullBlocks * perBlock;
  if (done < nElem) {
    const size_t rem = nElem - done;
    quant_tail<<<(unsigned)((rem + 255) / 256), 256, 0, stream>>>(
        x, q_out, Q_out, (unsigned long long)done, (unsigned long long)nElem);
  }
  copy_w<<<1, 64, 0, stream>>>(W, W_out, nW);
}